// customIdwtHaar_2d_43843026158255
// MI455X (gfx1250) — compile-verified
//
#include <hip/hip_runtime.h>
#include <cstdint>
#include <cstddef>

// Inverse 2D Haar synthesis for [4,8,512,512] subbands -> four [4,8,1024,1024]
// outputs concatenated. Pure streaming op (AI ~ 1 flop / 5 bytes).
// Key trick: with reflect padding, output rows {2r-1, 2r} BOTH map to input
// row r, so each block stages exactly one input row per subband via async
// global->LDS DMA -> each input byte crosses HBM exactly once.
// Traffic: 128 MiB read + 512 MiB NT-written = 640 MiB -> ~27us @ 23.3 TB/s.
// WMMA is not applicable (one multiply per output element).

typedef float v4f __attribute__((ext_vector_type(4)));

#define B_ 4
#define C_ 8
#define H_ 512
#define W_ 512
#define P_ (B_ * C_)                 // 32 planes
#define OH_ (2 * H_)
#define OW_ (2 * W_)
#define IN_PLANE (H_ * W_)           // 262144
#define OUT_PLANE (OH_ * OW_)        // 1048576
#define OUT_SUB ((size_t)P_ * (size_t)OUT_PLANE)  // elems per subband output

__global__ __launch_bounds__(256) void idwt_haar_kernel(
    const float* __restrict__ LL, const float* __restrict__ LH,
    const float* __restrict__ HL, const float* __restrict__ HH,
    float* __restrict__ out)
{
    // 4 subbands x 1 input row x 512 floats = 8 KB staged per block
    __shared__ __align__(16) float smem[4 * W_];

    const int t  = threadIdx.x;
    const int bx = blockIdx.x;
    const int p  = bx >> 9;              // plane 0..31
    const int r  = bx & 511;             // input row; serves output rows 2r-1, 2r

    const float* ins[4] = {LL, LH, HL, HH};

    // ---- Stage input row r of all 4 subbands into LDS (ASYNCcnt DMA path) ----
    // 512 x b128 transfers; 2 per thread.
#pragma unroll
    for (int it = 0; it < 2; ++it) {
        const int v = t + it * 256;          // vec4 tile id 0..511
        const int s = v >> 7;                // subband 0..3
        const int c = (v & 127) * 4;         // starting column (float index)
        const float* gp = ins[s] + (size_t)p * IN_PLANE + (size_t)r * W_ + c;
        // Low 32 bits of a generic LDS pointer are the LDS byte offset
        // (flat-LDS addressing truncates to addr[31:0]).
        const uint32_t la = (uint32_t)(uintptr_t)&smem[s * W_ + c];
        asm volatile("global_load_async_to_lds_b128 %0, %1, off"
                     :: "v"(la), "v"(gp) : "memory");
    }
    // Each wave waits for its own async transfers, then workgroup barrier.
    asm volatile("s_wait_asynccnt 0" ::: "memory");
    __syncthreads();

    // ---- Compute: out[i][j] = sign * 0.5 * x[rmap(i)][cmap(j)].
    // rs=0 -> output row i = 2r-1 (odd parity); rs=1 -> i = 2r (even parity).
    // cmap over an 8-wide group at j0 = g*8: cols c0,c0+1,c0+1,c0+2,c0+2,
    // c0+3,c0+3,min(c0+4,W-1) (reflect clamp on the right edge).
    const int rs = t >> 7;          // 0: odd output row, 1: even output row
    const int g  = t & 127;         // column group (8 outputs each)
    const int c0 = g * 4;
    const int c4 = (c0 + 4 < W_) ? (c0 + 4) : (W_ - 1);
    const int i  = 2 * r - 1 + rs;                 // -1 only when r==0, rs==0
    const bool valid = (i >= 0);
    const bool extra = (r == H_ - 1) && (rs == 1); // also emit row 2H-1
    float* outp  = out + (size_t)p * OUT_PLANE + (size_t)i * OW_ + (size_t)g * 8;
    float* outp2 = out + (size_t)p * OUT_PLANE + (size_t)(OH_ - 1) * OW_ + (size_t)g * 8;

    const v4f pos = {0.5f, 0.5f, 0.5f, 0.5f};
    const v4f alt = {0.5f, -0.5f, 0.5f, -0.5f};

#pragma unroll
    for (int s = 0; s < 4; ++s) {
        const float* rowp = &smem[s * W_];
        const v4f  a  = *(const v4f*)(rowp + c0);   // ds_load_b128, 16B aligned
        const float a4 = rowp[c4];

        const v4f lo = {a.x, a.y, a.y, a.z};
        const v4f hi = {a.z, a.w, a.w, a4};

        // Per-subband sign vectors for even / odd output rows:
        //   LL: +0.5 both | LH: even -, odd + | HL: col-alternating both
        //   HH: even +alt, odd -alt
        v4f sgE, sgO;
        if (s == 0)      { sgE = pos;  sgO = pos;  }
        else if (s == 1) { sgE = -pos; sgO = pos;  }
        else if (s == 2) { sgE = alt;  sgO = alt;  }
        else             { sgE = alt;  sgO = -alt; }
        const v4f sg = rs ? sgE : sgO;

        if (valid) {
            v4f* dst = (v4f*)(outp + (size_t)s * OUT_SUB);
            __builtin_nontemporal_store(lo * sg, dst);      // write-once: NT
            __builtin_nontemporal_store(hi * sg, dst + 1);
        }
        if (extra) {    // bottom reflect row 2H-1 (odd signs), same input row
            v4f* dst2 = (v4f*)(outp2 + (size_t)s * OUT_SUB);
            __builtin_nontemporal_store(lo * sgO, dst2);
            __builtin_nontemporal_store(hi * sgO, dst2 + 1);
        }
    }
}

extern "C" void kernel_launch(void* const* d_in, const int* in_sizes, int n_in,
                              void* d_out, int out_size, void* d_ws, size_t ws_size,
                              hipStream_t stream) {
    (void)in_sizes; (void)n_in; (void)d_ws; (void)ws_size; (void)out_size;
    const float* LL = (const float*)d_in[0];
    const float* LH = (const float*)d_in[1];
    const float* HL = (const float*)d_in[2];
    const float* HH = (const float*)d_in[3];
    // d_in[4..7] are the fixed +-0.5 Haar taps; folded into the kernel constants.
    float* out = (float*)d_out;

    const int blocks = P_ * H_;   // 32 planes * 512 input rows = 16384
    idwt_haar_kernel<<<blocks, 256, 0, stream>>>(LL, LH, HL, HH, out);
}